// FilterDetections_45827301048438
// MI455X (gfx1250) — compile-verified
//
#include <hip/hip_runtime.h>
#include <stdint.h>

// Problem constants (from setup_inputs / reference)
#define BATCH   8
#define NBOX    30000
#define NCLS    80
#define NP      30720            // NBOX padded to a multiple of 512; fits 16-bit tile_dim0
#define MAXDET  300
#define NMS_T   0.5f
#define SCORE_T 0.05f
#define NEG_INF (-__builtin_inff())

typedef uint32_t v4u  __attribute__((ext_vector_type(4)));
typedef int32_t  v8i_t __attribute__((ext_vector_type(8)));
typedef int32_t  v4i_t __attribute__((ext_vector_type(4)));

#if __has_builtin(__builtin_amdgcn_tensor_load_to_lds) && __has_builtin(__builtin_amdgcn_s_wait_tensorcnt)
#define USE_TDM 1
#endif

#ifdef USE_TDM
// Contiguous 1-D tile (nelem f32, nelem < 65536) global -> LDS via the Tensor
// Data Mover. Caller must be a single wave; completion via TENSORcnt.
__device__ __forceinline__ void tdm_load_1d(const float* gsrc, float* ldst, int nelem) {
  uint64_t ga  = (uint64_t)(uintptr_t)gsrc;
  uint32_t lds = (uint32_t)(uintptr_t)ldst;   // low 32 bits of generic addr == LDS byte offset
  // D# group0: count=1 | lds_addr | global_addr[56:0] | type=2
  v4u g0 = { 1u, lds, (uint32_t)ga,
             (uint32_t)((ga >> 32) & 0x1FFFFFFu) | (2u << 30) };
  // D# group1: data_size=4B; tensor_dim0=nelem, tensor_dim1=1; tile_dim0=nelem,
  // tile_dim1=1; tensor_dim0_stride=nelem
  uint32_t lo16 = ((uint32_t)nelem & 0xFFFFu) << 16;
  v8i_t g1 = { (int)0x20000, (int)lo16, (int)0x10000, (int)lo16, 1, nelem, 0, 0 };
  v4i_t gz = { 0, 0, 0, 0 };
  v8i_t g4 = { 0, 0, 0, 0, 0, 0, 0, 0 };      // groups 2/3 unused for <=2-D tensors
  __builtin_amdgcn_tensor_load_to_lds(g0, g1, gz, gz, g4, 0);   // TENSORcnt++
  __builtin_amdgcn_s_wait_tensorcnt(0);                         // s_wait_tensorcnt 0
}
#endif

// ---------------------------------------------------------------------------
// Kernel 1: canonicalize box geometry into packed float4 (ymin,ymax,xmin,xmax).
// One b128 store here; one coalesced b128 load per element in the NMS hot loop.
// Pad rows (n >= NBOX) are the zero/dummy box (area 0).
// ---------------------------------------------------------------------------
__global__ void geom_kernel(const float* __restrict__ boxes, float4* __restrict__ geom) {
  int t = blockIdx.x * blockDim.x + threadIdx.x;
  const int total = BATCH * NP;
  if (t >= total) return;
  int b = t / NP, n = t - b * NP;
  float4 g = { 0.f, 0.f, 0.f, 0.f };
  if (n < NBOX) {
    const float* p = boxes + ((size_t)b * NBOX + n) * 4;
    float a = p[0], c = p[1], d = p[2], e = p[3];
    g.x = fminf(a, d); g.y = fmaxf(a, d);   // ymin, ymax
    g.z = fminf(c, e); g.w = fmaxf(c, e);   // xmin, xmax
  }
  geom[t] = g;
}

// ---------------------------------------------------------------------------
// Kernel 2: tiled transpose classification [B,N,C] -> scoresT [B,C,NP].
// Pad (n >= NBOX) with -inf so the padded tail can never win an argmax
// (this is also the reference's dummy row).
// ---------------------------------------------------------------------------
__global__ void transpose_kernel(const float* __restrict__ cls, float* __restrict__ scoresT) {
  __shared__ float tile[32][33];
  const int b  = blockIdx.z;
  const int c0 = blockIdx.x * 32;
  const int n0 = blockIdx.y * 32;
  for (int j = threadIdx.y; j < 32; j += blockDim.y) {
    int n = n0 + j;
    int c = c0 + threadIdx.x;                         // coalesced along class dim
    float v = NEG_INF;
    if (n < NBOX && c < NCLS) v = cls[((size_t)b * NBOX + n) * NCLS + c];
    tile[j][threadIdx.x] = v;
  }
  __syncthreads();
  for (int j = threadIdx.y; j < 32; j += blockDim.y) {
    int c = c0 + j;
    int n = n0 + threadIdx.x;                         // coalesced along box dim
    if (c < NCLS && n < NP)
      scoresT[((size_t)b * NCLS + c) * NP + n] = tile[threadIdx.x][j];
  }
}

// argmax combine: max score, ties -> lower index (matches jnp.argmax)
__device__ __forceinline__ void amax2(float& s, int& i, float s2, int i2) {
  if (s2 > s || (s2 == s && i2 < i)) { s = s2; i = i2; }
}

// ---------------------------------------------------------------------------
// Kernel 3: one workgroup per (batch, class). Full padded score vector lives
// in LDS (120 KB — needs CDNA5's 320KB WGP LDS; 2 WGs/WGP). Filled via the
// Tensor Data Mover; 300 greedy argmax+suppress rounds entirely out of LDS/L2.
// ---------------------------------------------------------------------------
__global__ __launch_bounds__(512)
void nms_kernel(const float* __restrict__ scoresT, const float4* __restrict__ geom,
                float* __restrict__ sel_score, int* __restrict__ sel_idx) {
  extern __shared__ unsigned char smem_raw[];
  float* sc    = (float*)smem_raw;          // NP floats: live scores
  float* red_s = sc + NP;                   // 16 per-wave partial scores
  int*   red_i = (int*)(red_s + 16);        // 16 per-wave partial indices
  int*   win_v = red_i + 16;                // winner-valid flag
  float* wbox  = (float*)(win_v + 1);       // winner geometry: y0,y1,x0,x1,area

  const int tid  = threadIdx.x;
  const int lane = tid & 31;
  const int wave = tid >> 5;
  const int bc   = blockIdx.x;
  const int b    = bc / NCLS;

  const float4* gbox = geom + (size_t)b * NP;
  const float*  src  = scoresT + (size_t)bc * NP;

  // ---- fill LDS score vector (TDM async DMA, issued by wave 0) ----
#ifdef USE_TDM
  if (tid < 32) tdm_load_1d(src, sc, NP);
#else
  for (int j = tid; j < NP; j += 512) sc[j] = src[j];
#endif
  __syncthreads();

  for (int t = 0; t < MAXDET; ++t) {
    // per-thread argmax over strided chunk (60 LDS reads)
    float bs = NEG_INF; int bi = NP - 1;
    #pragma unroll 4
    for (int k = 0; k < NP / 512; ++k) {
      int j = tid + (k << 9);
      float s = sc[j];
      if (s > bs) { bs = s; bi = j; }
    }
    // wave32 shuffle reduction
    for (int m = 16; m; m >>= 1)
      amax2(bs, bi, __shfl_xor(bs, m, 32), __shfl_xor(bi, m, 32));
    if (lane == 0) { red_s[wave] = bs; red_i[wave] = bi; }
    __syncthreads();
    if (tid < 32) {
      float s = (lane < 16) ? red_s[lane] : NEG_INF;
      int   i = (lane < 16) ? red_i[lane] : NP - 1;
      for (int m = 8; m; m >>= 1)
        amax2(s, i, __shfl_xor(s, m, 32), __shfl_xor(i, m, 32));
      if (lane == 0) {
        int valid = (s > SCORE_T);
        *win_v = valid;
        if (valid) {
          float4 g = gbox[i];
          wbox[0] = g.x; wbox[1] = g.y; wbox[2] = g.z; wbox[3] = g.w;
          wbox[4] = (g.y - g.x) * (g.w - g.z);           // winner area
          sc[i] = NEG_INF;                               // remove winner
          sel_score[(size_t)bc * MAXDET + t] = s;
          sel_idx  [(size_t)bc * MAXDET + t] = i;        // i < NBOX since pad scores are -inf
        }
      }
    }
    __syncthreads();
    if (!*win_v) {            // uniform: pad the rest with the dummy and stop
      for (int tt = t + tid; tt < MAXDET; tt += 512) {
        sel_score[(size_t)bc * MAXDET + tt] = NEG_INF;
        sel_idx  [(size_t)bc * MAXDET + tt] = NBOX;
      }
      break;
    }
    // suppression: IoU(winner, j) > 0.5  <=>  inter > 0.5*union (no divide)
    const float wy0 = wbox[0], wy1 = wbox[1], wx0 = wbox[2], wx1 = wbox[3], war = wbox[4];
    #pragma unroll 4
    for (int k = 0; k < NP / 512; ++k) {
      int j = tid + (k << 9);
      float s = sc[j];
      if (s == NEG_INF) continue;                        // already dead: skip geometry load
      if (k + 1 < NP / 512) __builtin_prefetch(&gbox[j + 512], 0, 1);
      float4 g = gbox[j];                                // one coalesced b128 load
      float ih = fminf(wy1, g.y) - fmaxf(wy0, g.x);
      float iw = fminf(wx1, g.w) - fmaxf(wx0, g.z);
      ih = fmaxf(ih, 0.f); iw = fmaxf(iw, 0.f);
      float inter = ih * iw;
      if (inter > 0.f) {
        float areaj = (g.y - g.x) * (g.w - g.z);
        float uni = war + areaj - inter;
        if (inter > NMS_T * uni) sc[j] = NEG_INF;
      }
    }
    __syncthreads();
  }
}

// ---------------------------------------------------------------------------
// Kernel 4: per-batch top-300 over the 80*300 NMS survivors (96KB in LDS,
// TDM-filled), gather boxes/labels, write the (boxes, scores, labels) tuple.
// ---------------------------------------------------------------------------
__global__ __launch_bounds__(512)
void topk_kernel(const float* __restrict__ sel_score, const int* __restrict__ sel_idx,
                 const float* __restrict__ boxes,
                 float* __restrict__ out_boxes, float* __restrict__ out_scores,
                 int* __restrict__ out_labels) {
  extern __shared__ unsigned char smem_raw[];
  const int M = NCLS * MAXDET;               // 24000 candidates
  float* cs    = (float*)smem_raw;           // M floats
  float* red_s = cs + M;
  int*   red_i = (int*)(red_s + 16);

  const int tid  = threadIdx.x;
  const int lane = tid & 31;
  const int wave = tid >> 5;
  const int b    = blockIdx.x;

  const float* src = sel_score + (size_t)b * M;
#ifdef USE_TDM
  if (tid < 32) tdm_load_1d(src, cs, M);
#else
  for (int j = tid; j < M; j += 512) cs[j] = src[j];
#endif
  __syncthreads();

  for (int t = 0; t < MAXDET; ++t) {
    float bs = NEG_INF; int bi = M - 1;
    for (int j = tid; j < M; j += 512) {
      float s = cs[j];
      if (s > bs || (s == bs && j < bi)) { bs = s; bi = j; }
    }
    for (int m = 16; m; m >>= 1)
      amax2(bs, bi, __shfl_xor(bs, m, 32), __shfl_xor(bi, m, 32));
    if (lane == 0) { red_s[wave] = bs; red_i[wave] = bi; }
    __syncthreads();
    if (tid < 32) {
      float s = (lane < 16) ? red_s[lane] : NEG_INF;
      int   i = (lane < 16) ? red_i[lane] : M - 1;
      for (int m = 8; m; m >>= 1)
        amax2(s, i, __shfl_xor(s, m, 32), __shfl_xor(i, m, 32));
      if (lane == 0) {
        int valid  = (s > SCORE_T);
        int n      = valid ? sel_idx[(size_t)b * M + i] : NBOX;
        int cls_id = i / MAXDET;
        float* ob  = out_boxes + ((size_t)b * MAXDET + t) * 4;
        if (valid && n < NBOX) {
          const float* p = boxes + ((size_t)b * NBOX + n) * 4;
          ob[0] = p[0]; ob[1] = p[1]; ob[2] = p[2]; ob[3] = p[3];
          out_labels[b * MAXDET + t] = cls_id;
        } else {
          ob[0] = ob[1] = ob[2] = ob[3] = 0.f;
          out_labels[b * MAXDET + t] = -1;
        }
        out_scores[b * MAXDET + t] = valid ? s : NEG_INF;
        if (valid) cs[i] = NEG_INF;
      }
    }
    __syncthreads();
  }
}

// ---------------------------------------------------------------------------
extern "C" void kernel_launch(void* const* d_in, const int* in_sizes, int n_in,
                              void* d_out, int out_size, void* d_ws, size_t ws_size,
                              hipStream_t stream) {
  const float* boxes = (const float*)d_in[0];   // [8,30000,4]
  const float* cls   = (const float*)d_in[1];   // [8,30000,80]

  // workspace layout (~84 MB): scoresT | geom(float4) | sel_score | sel_idx
  float* ws = (float*)d_ws;
  const size_t scoresT_sz = (size_t)BATCH * NCLS * NP;     // 19,660,800 f
  const size_t geom_sz    = (size_t)4 * BATCH * NP;        //    983,040 f
  const size_t sel_sz     = (size_t)BATCH * NCLS * MAXDET; //    192,000
  float*  scoresT   = ws;
  float4* geom      = (float4*)(scoresT + scoresT_sz);
  float*  sel_score = scoresT + scoresT_sz + geom_sz;
  int*    sel_idx   = (int*)(sel_score + sel_sz);
  (void)ws_size; (void)in_sizes; (void)n_in; (void)out_size;

  // output tuple, flattened in return order: boxes [8,300,4], scores [8,300], labels [8,300]
  float* out_boxes  = (float*)d_out;
  float* out_scores = out_boxes + (size_t)BATCH * MAXDET * 4;
  int*   out_labels = (int*)(out_scores + (size_t)BATCH * MAXDET);

  geom_kernel<<<(BATCH * NP + 255) / 256, 256, 0, stream>>>(boxes, geom);

  dim3 tgrid((NCLS + 31) / 32, NP / 32, BATCH);
  transpose_kernel<<<tgrid, dim3(32, 8), 0, stream>>>(cls, scoresT);

  const size_t nms_smem = (size_t)NP * sizeof(float) + 256;             // ~120.2 KB
  nms_kernel<<<BATCH * NCLS, 512, nms_smem, stream>>>(scoresT, geom, sel_score, sel_idx);

  const size_t topk_smem = (size_t)NCLS * MAXDET * sizeof(float) + 256; // ~94 KB
  topk_kernel<<<BATCH, 512, topk_smem, stream>>>(sel_score, sel_idx, boxes,
                                                 out_boxes, out_scores, out_labels);
}